// RelativeAttentionLayer_6914897347235
// MI455X (gfx1250) — compile-verified
//
#include <hip/hip_runtime.h>
#include <math.h>
#include <stdint.h>

#define BB 2
#define LL 512
#define CC 256
#define HH 8
#define DD 32

typedef float v2f __attribute__((ext_vector_type(2)));
typedef float v8f __attribute__((ext_vector_type(8)));

__device__ __forceinline__ v8f wmma4(v2f a, v2f b, v8f c) {
  // V_WMMA_F32_16X16X4_F32 : D = A(16x4) * B(4x16) + C(16x16)
  return __builtin_amdgcn_wmma_f32_16x16x4_f32(
      /*neg_a=*/false, a, /*neg_b=*/false, b,
      /*c_mod=*/(short)0, c, /*reuse_a=*/false, /*reuse_b=*/false);
}

__device__ __forceinline__ v2f gld2(const float* p) { return *(const v2f*)p; }

// ---------------------------------------------------------------------------
// 1) QKV projection: [B*L, C] x [C, 3C] -> q/k/v in [B,H,L,D]
// ---------------------------------------------------------------------------
__global__ __launch_bounds__(32) void qkv_kernel(const float* __restrict__ x,
                                                 const float* __restrict__ w_qkv,
                                                 float* __restrict__ qw,
                                                 float* __restrict__ kw,
                                                 float* __restrict__ vw) {
  const int lane = threadIdx.x & 31;
  const int lrow = lane & 15;
  const int lhi  = lane >> 4;
  const int mt = blockIdx.x;   // 0..63   (M = B*L = 1024)
  const int nt = blockIdx.y;   // 0..47   (N = 3C = 768)
  const float* a0 = x     + (size_t)(mt * 16 + lrow) * CC + 2 * lhi;
  const float* b0 = w_qkv + (size_t)(nt * 16 + lrow) * CC + 2 * lhi; // B[k][n]=w_qkv[n][k]
  v8f acc = {0.f,0.f,0.f,0.f,0.f,0.f,0.f,0.f};
#pragma unroll 8
  for (int k0 = 0; k0 < CC; k0 += 4)
    acc = wmma4(gld2(a0 + k0), gld2(b0 + k0), acc);

  const int n = nt * 16 + lrow;
  const int which = n >> 8;          // 0=q 1=k 2=v
  const int h = (n & 255) >> 5;
  const int d = n & 31;
  float* dst = (which == 0) ? qw : ((which == 1) ? kw : vw);
#pragma unroll
  for (int r = 0; r < 8; ++r) {
    const int row = mt * 16 + r + 8 * lhi;   // flat b*L + l
    const int b = row >> 9, l = row & (LL - 1);
    dst[(((size_t)b * HH + h) * LL + l) * DD + d] = acc[r];
  }
}

// ---------------------------------------------------------------------------
// 2) Relative bias: rel = LN_h(gelu(relative_pos @ w_rel^T + b_rel)) -> [B,H,L,L]
//    THE bandwidth kernel: streams 536 MB of relative_pos.
//    Uses gfx1250 async memory->LDS copies (ASYNCcnt) with double buffering:
//    chunk c+1 is in flight while WMMAs consume chunk c from LDS.
// ---------------------------------------------------------------------------
__global__ __launch_bounds__(256) void rel_kernel(const float* __restrict__ rp,
                                                  const float* __restrict__ w_rel,
                                                  const float* __restrict__ b_rel,
                                                  const float* __restrict__ gam,
                                                  const float* __restrict__ bet,
                                                  float* __restrict__ out) {
  // stride 36 floats = 144 B: keeps 16B alignment for async b128 writes and
  // maps the 16 wmma rows (stride-36) onto 16 distinct LDS banks.
  __shared__ float lds[2][128][36];
  const int tid  = threadIdx.x;
  const int wave = tid >> 5;
  const int lane = tid & 31;
  const int lrow = lane & 15;
  const int lhi  = lane >> 4;
  const int hq   = lrow & 7;
  const long long rowBase = (long long)blockIdx.x * 128;

  // each thread owns 4 x 16B segments per 128x32 chunk
  const int ld_row0 = tid >> 3;        // rows tid/8, +32, +64, +96
  const int ld_seg  = (tid & 7) * 4;   // float offset within 32-wide chunk

  v8f acc = {0.f,0.f,0.f,0.f,0.f,0.f,0.f,0.f};

  // issue chunk 0 into buffer 0
#pragma unroll
  for (int it = 0; it < 4; ++it) {
    const int row = ld_row0 + it * 32;
    const unsigned ldsOff =
        (unsigned)(uintptr_t)&lds[0][row][ld_seg];
    const unsigned long long ga =
        (unsigned long long)(uintptr_t)(rp + (rowBase + row) * CC + ld_seg);
    asm volatile("global_load_async_to_lds_b128 %0, %1, off"
                 :: "v"(ldsOff), "v"(ga) : "memory");
  }

  for (int c = 0; c < CC / 32; ++c) {          // 8 chunks of K=32
    const int k0 = c * 32;
    if (c + 1 < CC / 32) {
      // prefetch next chunk into the other buffer (stays in flight)
#pragma unroll
      for (int it = 0; it < 4; ++it) {
        const int row = ld_row0 + it * 32;
        const unsigned ldsOff =
            (unsigned)(uintptr_t)&lds[(c + 1) & 1][row][ld_seg];
        const unsigned long long ga =
            (unsigned long long)(uintptr_t)(rp + (rowBase + row) * CC + k0 + 32 + ld_seg);
        asm volatile("global_load_async_to_lds_b128 %0, %1, off"
                     :: "v"(ldsOff), "v"(ga) : "memory");
      }
      // async ops complete in order: <=4 outstanding => current chunk landed
      asm volatile("s_wait_asynccnt 0x4" ::: "memory");
    } else {
      asm volatile("s_wait_asynccnt 0x0" ::: "memory");
    }
    __syncthreads();

    const float* wb = w_rel + (size_t)hq * CC + k0 + 2 * lhi;
#pragma unroll
    for (int kk = 0; kk < 32; kk += 4) {
      v2f a = *(const v2f*)&lds[c & 1][wave * 16 + lrow][kk + 2 * lhi];
      v2f bv = gld2(wb + kk);
      if (lrow >= 8) { bv.x = 0.f; bv.y = 0.f; }  // pad heads 8..15 with zeros
      acc = wmma4(a, bv, acc);
    }
    __syncthreads();   // all waves done reading before buffer is overwritten
  }

  const float bias = b_rel[hq];
  const float gg   = gam[hq];
  const float bb   = bet[hq];
  float vals[8];
#pragma unroll
  for (int r = 0; r < 8; ++r) {
    float xv = acc[r] + bias;
    xv = 0.5f * xv * (1.0f + erff(xv * 0.70710678118654752f));   // exact GELU
    // LayerNorm over heads: heads 0..7 of row M live in an 8-lane group
    float s = xv;
#pragma unroll
    for (int o = 1; o < 8; o <<= 1) s += __shfl_xor(s, o, 8);
    const float mu = s * 0.125f;
    const float d  = xv - mu;
    float v2 = d * d;
#pragma unroll
    for (int o = 1; o < 8; o <<= 1) v2 += __shfl_xor(v2, o, 8);
    vals[r] = d * rsqrtf(v2 * 0.125f + 1e-5f) * gg + bb;
  }
  if (lrow < 8) {
#pragma unroll
    for (int r = 0; r < 8; ++r) {
      const long long row = rowBase + wave * 16 + r + 8 * lhi;  // flat (b*L + l)*L + m
      const int b   = (int)(row >> 18);
      const int rem = (int)(row & ((1 << 18) - 1));
      const int l = rem >> 9, m = rem & 511;
      out[(((long long)b * HH + lrow) * LL + l) * (long long)LL + m] = vals[r];
    }
  }
}

// ---------------------------------------------------------------------------
// 3) Scores: S = (q k^T + rel) * scale  (in place over the rel buffer)
// ---------------------------------------------------------------------------
__global__ __launch_bounds__(32) void scores_kernel(const float* __restrict__ qw,
                                                    const float* __restrict__ kw,
                                                    float* __restrict__ sc) {
  const int lane = threadIdx.x & 31;
  const int lrow = lane & 15;
  const int lhi  = lane >> 4;
  const int bh = blockIdx.x;   // 0..15
  const int lt = blockIdx.y;   // 0..31
  const int mt = blockIdx.z;   // 0..31
  const float* a0 = qw + (size_t)bh * LL * DD + (lt * 16 + lrow) * DD + 2 * lhi;
  const float* b0 = kw + (size_t)bh * LL * DD + (mt * 16 + lrow) * DD + 2 * lhi;
  v8f acc = {0.f,0.f,0.f,0.f,0.f,0.f,0.f,0.f};
#pragma unroll
  for (int k0 = 0; k0 < DD; k0 += 4)
    acc = wmma4(gld2(a0 + k0), gld2(b0 + k0), acc);

  float* sb = sc + (size_t)bh * LL * LL;
  const float scale = 0.17677669529663689f;   // D^-0.5
#pragma unroll
  for (int r = 0; r < 8; ++r) {
    const int l = lt * 16 + r + 8 * lhi;
    const int m = mt * 16 + lrow;
    const size_t idx = (size_t)l * LL + m;
    sb[idx] = (acc[r] + sb[idx]) * scale;   // bias pre-scale, matching reference
  }
}

// ---------------------------------------------------------------------------
// 4) Row softmax over L=512 (one row per wave)
// ---------------------------------------------------------------------------
__global__ __launch_bounds__(256) void softmax_kernel(float* __restrict__ sc) {
  const int lane = threadIdx.x & 31;
  const int wave = threadIdx.x >> 5;
  float* p = sc + ((size_t)blockIdx.x * 8 + wave) * LL;
  float x[16];
  float mx = -3.402823466e38f;
#pragma unroll
  for (int i = 0; i < 16; ++i) { x[i] = p[lane + i * 32]; mx = fmaxf(mx, x[i]); }
#pragma unroll
  for (int o = 16; o > 0; o >>= 1) mx = fmaxf(mx, __shfl_xor(mx, o, 32));
  float s = 0.f;
#pragma unroll
  for (int i = 0; i < 16; ++i) { x[i] = __expf(x[i] - mx); s += x[i]; }
#pragma unroll
  for (int o = 16; o > 0; o >>= 1) s += __shfl_xor(s, o, 32);
  const float inv = 1.0f / s;
#pragma unroll
  for (int i = 0; i < 16; ++i) p[lane + i * 32] = x[i] * inv;
}

// ---------------------------------------------------------------------------
// 5) AV: out = attn @ v, written as [B,L,C] for the projection
// ---------------------------------------------------------------------------
__global__ __launch_bounds__(32) void av_kernel(const float* __restrict__ sc,
                                                const float* __restrict__ vw,
                                                float* __restrict__ ao) {
  const int lane = threadIdx.x & 31;
  const int lrow = lane & 15;
  const int lhi  = lane >> 4;
  const int bh = blockIdx.x;   // 0..15
  const int lt = blockIdx.y;   // 0..31
  const int dt = blockIdx.z;   // 0..1
  const float* ab = sc + (size_t)bh * LL * LL + (lt * 16 + lrow) * LL + 2 * lhi;
  const float* vb = vw + (size_t)bh * LL * DD + (size_t)(2 * lhi) * DD + dt * 16 + lrow;
  v8f acc = {0.f,0.f,0.f,0.f,0.f,0.f,0.f,0.f};
#pragma unroll 8
  for (int k0 = 0; k0 < LL; k0 += 4) {
    v2f a = gld2(ab + k0);
    v2f b;
    b.x = vb[(size_t)k0 * DD];        // v[k0 + 2*lhi    ][d]
    b.y = vb[(size_t)k0 * DD + DD];   // v[k0 + 2*lhi + 1][d]
    acc = wmma4(a, b, acc);
  }
  const int b_ = bh >> 3, h = bh & 7;
#pragma unroll
  for (int r = 0; r < 8; ++r) {
    const int l = lt * 16 + r + 8 * lhi;
    ao[((size_t)(b_ * LL + l)) * CC + h * DD + dt * 16 + lrow] = acc[r];
  }
}

// ---------------------------------------------------------------------------
// 6) Output projection: [B*L, C] x [C, C] + b_proj
// ---------------------------------------------------------------------------
__global__ __launch_bounds__(32) void proj_kernel(const float* __restrict__ ao,
                                                  const float* __restrict__ wp,
                                                  const float* __restrict__ bp,
                                                  float* __restrict__ out) {
  const int lane = threadIdx.x & 31;
  const int lrow = lane & 15;
  const int lhi  = lane >> 4;
  const int mt = blockIdx.x;   // 0..63
  const int nt = blockIdx.y;   // 0..15
  const float* a0 = ao + (size_t)(mt * 16 + lrow) * CC + 2 * lhi;
  const float* b0 = wp + (size_t)(nt * 16 + lrow) * CC + 2 * lhi;
  v8f acc = {0.f,0.f,0.f,0.f,0.f,0.f,0.f,0.f};
#pragma unroll 8
  for (int k0 = 0; k0 < CC; k0 += 4)
    acc = wmma4(gld2(a0 + k0), gld2(b0 + k0), acc);
  const float bias = bp[nt * 16 + lrow];
#pragma unroll
  for (int r = 0; r < 8; ++r)
    out[(size_t)(mt * 16 + r + 8 * lhi) * CC + nt * 16 + lrow] = acc[r] + bias;
}

// ---------------------------------------------------------------------------
extern "C" void kernel_launch(void* const* d_in, const int* in_sizes, int n_in,
                              void* d_out, int out_size, void* d_ws, size_t ws_size,
                              hipStream_t stream) {
  const float* x      = (const float*)d_in[0];
  const float* rp     = (const float*)d_in[1];
  const float* w_qkv  = (const float*)d_in[2];
  const float* w_rel  = (const float*)d_in[3];
  const float* b_rel  = (const float*)d_in[4];
  const float* ln_g   = (const float*)d_in[5];
  const float* ln_b   = (const float*)d_in[6];
  const float* w_proj = (const float*)d_in[7];
  const float* b_proj = (const float*)d_in[8];
  float* out = (float*)d_out;

  float* ws = (float*)d_ws;
  const size_t qkv_sz = (size_t)BB * HH * LL * DD;   // 262144 floats each
  float* qw = ws;
  float* kw = qw + qkv_sz;
  float* vw = kw + qkv_sz;
  float* sc = vw + qkv_sz;                           // [B,H,L,L] = 4M floats
  float* ao = sc + (size_t)BB * HH * LL * LL;        // [B,L,C]

  qkv_kernel   <<<dim3(64, 48),     32,  0, stream>>>(x, w_qkv, qw, kw, vw);
  rel_kernel   <<<4096,             256, 0, stream>>>(rp, w_rel, b_rel, ln_g, ln_b, sc);
  scores_kernel<<<dim3(16, 32, 32), 32,  0, stream>>>(qw, kw, sc);
  softmax_kernel<<<1024,            256, 0, stream>>>(sc);
  av_kernel    <<<dim3(16, 32, 2),  32,  0, stream>>>(sc, vw, ao);
  proj_kernel  <<<dim3(64, 16),     32,  0, stream>>>(ao, w_proj, b_proj, out);
}